// MPLayer_13700945674315
// MI455X (gfx1250) — compile-verified
//
#include <hip/hip_runtime.h>
#include <stdint.h>

#define TPB 256
#define EPT 8   // edges per thread: two b128 loads per stream per thread

typedef int v4i __attribute__((ext_vector_type(4)));

__global__ void mp_zero_kernel(float* __restrict__ out, int n) {
    int i = blockIdx.x * blockDim.x + threadIdx.x;
    if (i < n) out[i] = 0.0f;
}

// E_CONST != 0 -> compile-time divisor (magic-multiply mod, no v_div sequence)
template <unsigned E_CONST>
__global__ __launch_bounds__(TPB) void mp_scatter_kernel(
    const int*   __restrict__ row,    // [nnz]
    const int*   __restrict__ col,    // [nnz]
    const int*   __restrict__ mask,   // [nnz]
    const float* __restrict__ e,      // [E]
    const float* __restrict__ p,      // [R]
    float*       __restrict__ out,    // [E]
    int nnz, unsigned num_entities, int num_rel)
{
    const unsigned E = E_CONST ? E_CONST : num_entities;

    // Stage p_scores (<=1KB) into LDS via the CDNA5 async global->LDS path.
    __shared__ float s_p[256];
    const unsigned tid = threadIdx.x;
    if (tid < (unsigned)num_rel && tid < 256u) {
        // low 32 bits of a generic pointer into the LDS aperture = LDS byte offset
        unsigned lds_addr = (unsigned)(uintptr_t)(&s_p[0]) + tid * 4u;
        unsigned goff     = tid * 4u;  // byte offset from base SGPR pair (GVS mode)
        asm volatile("global_load_async_to_lds_b32 %0, %1, %2"
                     :
                     : "v"(lds_addr), "v"(goff), "s"(p)
                     : "memory");
    }
    asm volatile("s_wait_asynccnt 0x0" ::: "memory");
    __syncthreads();

    const long base = ((long)blockIdx.x * TPB + tid) * EPT;
    if (base + EPT <= (long)nnz) {
        // One-pass streams: non-temporal so they don't evict e_scores/out from L2.
        v4i r0 = __builtin_nontemporal_load((const v4i*)(row  + base));
        v4i r1 = __builtin_nontemporal_load((const v4i*)(row  + base) + 1);
        v4i c0 = __builtin_nontemporal_load((const v4i*)(col  + base));
        v4i c1 = __builtin_nontemporal_load((const v4i*)(col  + base) + 1);
        v4i m0 = __builtin_nontemporal_load((const v4i*)(mask + base));
        v4i m1 = __builtin_nontemporal_load((const v4i*)(mask + base) + 1);
#pragma unroll
        for (int k = 0; k < 4; ++k) {
            float contrib = s_p[m0[k]] * e[r0[k]];
            unsigned seg  = (unsigned)c0[k] % E;   // magic-mul mod when E_CONST
            atomicAdd(&out[seg], contrib);         // global_atomic_add_f32 (no return)
        }
#pragma unroll
        for (int k = 0; k < 4; ++k) {
            float contrib = s_p[m1[k]] * e[r1[k]];
            unsigned seg  = (unsigned)c1[k] % E;
            atomicAdd(&out[seg], contrib);
        }
    } else {
        for (int k = 0; k < EPT; ++k) {
            long i = base + k;
            if (i < (long)nnz) {
                float contrib = s_p[mask[i]] * e[row[i]];
                unsigned seg  = (unsigned)col[i] % E;
                atomicAdd(&out[seg], contrib);
            }
        }
    }
}

extern "C" void kernel_launch(void* const* d_in, const int* in_sizes, int n_in,
                              void* d_out, int out_size, void* d_ws, size_t ws_size,
                              hipStream_t stream) {
    // setup_inputs order: indices [2,nnz], e_scores [E], p_scores [R], relation_mask [nnz]
    const int*   indices = (const int*)d_in[0];
    const float* e       = (const float*)d_in[1];
    const float* p       = (const float*)d_in[2];
    const int*   mask    = (const int*)d_in[3];
    float*       out     = (float*)d_out;

    const int nnz = in_sizes[3];
    const int E   = in_sizes[1];
    const int R   = in_sizes[2];

    const int* row = indices;        // indices[0, :]
    const int* col = indices + nnz;  // indices[1, :]

    // Zero the output (harness poisons d_out; atomics accumulate on top).
    int zb = (out_size + TPB - 1) / TPB;
    mp_zero_kernel<<<zb, TPB, 0, stream>>>(out, out_size);

    int sb = (nnz + TPB * EPT - 1) / (TPB * EPT);
    if (E == 100000) {
        mp_scatter_kernel<100000u><<<sb, TPB, 0, stream>>>(row, col, mask, e, p, out,
                                                           nnz, (unsigned)E, R);
    } else {
        mp_scatter_kernel<0u><<<sb, TPB, 0, stream>>>(row, col, mask, e, p, out,
                                                      nnz, (unsigned)E, R);
    }
}